// LSEP_Loss_8993661518527
// MI455X (gfx1250) — compile-verified
//
#include <hip/hip_runtime.h>
#include <hip/hip_bf16.h>

// LSEP loss: per-row masked logsumexp pair + softplus, mean over rows.
// B = 4096 rows, C = 8192 columns. Memory-bound: 256 MB single read ~= 11 us
// at 23.3 TB/s. Strategy: one block per row, stage the whole row (x + targets,
// 64 KB) into LDS with CDNA5 async global->LDS b128 copies, then two passes
// (max, sum of exp2) from LDS so global memory is read exactly once.

#define TPB       256
#define CDIM      8192
#define VEC_ITERS (CDIM / (TPB * 4))   // 8 float4/int4 chunks per thread

// ---- CDNA5 async global->LDS path (probe-confirmed AVAILABLE on this toolchain)
#if defined(__HIP_DEVICE_COMPILE__)
#if __has_builtin(__builtin_amdgcn_global_load_async_to_lds_b128)
#define HAVE_ASYNC_LDS 1
#else
#define HAVE_ASYNC_LDS 0
#warning "CDNA5 probe: __builtin_amdgcn_global_load_async_to_lds_b128 MISSING"
#endif
#if __has_builtin(__builtin_amdgcn_s_wait_asynccnt)
#define HAVE_WAIT_ASYNC 1
#else
#define HAVE_WAIT_ASYNC 0
#warning "CDNA5 probe: __builtin_amdgcn_s_wait_asynccnt MISSING"
#endif
#else
#define HAVE_ASYNC_LDS 0
#define HAVE_WAIT_ASYNC 0
#endif

// Builtin signature (from clang diagnostic): first param is
// 'int __attribute__((vector_size(16))) __device__ *' (AS1), second is the
// LDS-side pointer (AS3), then imm offset, imm cpol.
typedef int v4i_t __attribute__((vector_size(16)));
typedef __attribute__((address_space(1))) v4i_t as1_v4i;
typedef __attribute__((address_space(3))) v4i_t as3_v4i;

__device__ __forceinline__ void async_copy_b128(const void* g, void* l) {
#if HAVE_ASYNC_LDS
  // (global src, lds dst, imm offset, imm cpol)
  __builtin_amdgcn_global_load_async_to_lds_b128((as1_v4i*)g, (as3_v4i*)l, 0, 0);
#else
  *(float4*)l = *(const float4*)g;   // fallback: stage through VGPRs
#endif
}

__device__ __forceinline__ void wait_async_all() {
#if HAVE_ASYNC_LDS
#if HAVE_WAIT_ASYNC
  __builtin_amdgcn_s_wait_asynccnt(0);
#else
  asm volatile("s_wait_asynccnt 0" ::: "memory");
#endif
#endif
}

// ---- wave32 block reductions (8 waves -> LDS scratch -> wave 0) ----
__device__ __forceinline__ float block_reduce_max(float v, float* scr) {
#pragma unroll
  for (int off = 16; off > 0; off >>= 1)
    v = fmaxf(v, __shfl_xor(v, off, 32));
  const int lane = threadIdx.x & 31;
  const int wid  = threadIdx.x >> 5;
  if (lane == 0) scr[wid] = v;
  __syncthreads();
  if (threadIdx.x < 32) {
    float t = (lane < (TPB / 32)) ? scr[lane] : -__builtin_inff();
#pragma unroll
    for (int off = 4; off > 0; off >>= 1)
      t = fmaxf(t, __shfl_xor(t, off, 32));
    if (lane == 0) scr[0] = t;
  }
  __syncthreads();
  float r = scr[0];
  __syncthreads();   // scratch reuse safety
  return r;
}

__device__ __forceinline__ float block_reduce_sum(float v, float* scr) {
#pragma unroll
  for (int off = 16; off > 0; off >>= 1)
    v += __shfl_xor(v, off, 32);
  const int lane = threadIdx.x & 31;
  const int wid  = threadIdx.x >> 5;
  if (lane == 0) scr[wid] = v;
  __syncthreads();
  if (threadIdx.x < 32) {
    float t = (lane < (TPB / 32)) ? scr[lane] : 0.0f;
#pragma unroll
    for (int off = 4; off > 0; off >>= 1)
      t += __shfl_xor(t, off, 32);
    if (lane == 0) scr[0] = t;
  }
  __syncthreads();
  float r = scr[0];
  __syncthreads();
  return r;
}

// ---- per-row kernel: one block per row ----
__global__ __launch_bounds__(TPB) void lsep_row_kernel(
    const float* __restrict__ x, const int* __restrict__ tg,
    float* __restrict__ row_loss) {
  __shared__ __align__(16) float lx[CDIM];   // 32 KB
  __shared__ __align__(16) int   lt[CDIM];   // 32 KB
  __shared__ float scr[TPB / 32 * 2];

  const int       tid = threadIdx.x;
  const long long row = blockIdx.x;
  const float* gx = x  + row * (long long)CDIM;
  const int*   gt = tg + row * (long long)CDIM;

  // Stage full row into LDS: 16 async b128 copies per thread.
#pragma unroll
  for (int k = 0; k < VEC_ITERS; ++k) {
    const int e = (tid + k * TPB) * 4;
    async_copy_b128(gx + e, &lx[e]);
    async_copy_b128(gt + e, &lt[e]);
  }
  wait_async_all();
  __syncthreads();

  const float LOG2E = 1.4426950408889634f;
  const float NEGINF = -__builtin_inff();

  // Pass A: per-set maxima in base-2 domain (y = x * log2 e).
  float mn = NEGINF;   // max over negatives of  y
  float mp = NEGINF;   // max over positives of -y
#pragma unroll
  for (int k = 0; k < VEC_ITERS; ++k) {
    const int e = (tid + k * TPB) * 4;
    const float4 v = *(const float4*)(&lx[e]);
    const int4   t = *(const int4*)(&lt[e]);
    const float y0 = v.x * LOG2E, y1 = v.y * LOG2E;
    const float y2 = v.z * LOG2E, y3 = v.w * LOG2E;
    mn = fmaxf(mn, (t.x == 0) ?  y0 : NEGINF);
    mp = fmaxf(mp, (t.x != 0) ? -y0 : NEGINF);
    mn = fmaxf(mn, (t.y == 0) ?  y1 : NEGINF);
    mp = fmaxf(mp, (t.y != 0) ? -y1 : NEGINF);
    mn = fmaxf(mn, (t.z == 0) ?  y2 : NEGINF);
    mp = fmaxf(mp, (t.z != 0) ? -y2 : NEGINF);
    mn = fmaxf(mn, (t.w == 0) ?  y3 : NEGINF);
    mp = fmaxf(mp, (t.w != 0) ? -y3 : NEGINF);
  }
  mn = block_reduce_max(mn, scr);
  mp = block_reduce_max(mp, scr);

  // Pass B: exactly one exp2 per element (argument selected by mask).
  float sn = 0.0f, sp = 0.0f;
#pragma unroll
  for (int k = 0; k < VEC_ITERS; ++k) {
    const int e = (tid + k * TPB) * 4;
    const float4 v = *(const float4*)(&lx[e]);
    const int4   t = *(const int4*)(&lt[e]);
    {
      const float y = v.x * LOG2E; const bool neg = (t.x == 0);
      const float ex = exp2f(neg ? (y - mn) : (-y - mp));
      sn += neg ? ex : 0.0f; sp += neg ? 0.0f : ex;
    }
    {
      const float y = v.y * LOG2E; const bool neg = (t.y == 0);
      const float ex = exp2f(neg ? (y - mn) : (-y - mp));
      sn += neg ? ex : 0.0f; sp += neg ? 0.0f : ex;
    }
    {
      const float y = v.z * LOG2E; const bool neg = (t.z == 0);
      const float ex = exp2f(neg ? (y - mn) : (-y - mp));
      sn += neg ? ex : 0.0f; sp += neg ? 0.0f : ex;
    }
    {
      const float y = v.w * LOG2E; const bool neg = (t.w == 0);
      const float ex = exp2f(neg ? (y - mn) : (-y - mp));
      sn += neg ? ex : 0.0f; sp += neg ? 0.0f : ex;
    }
  }
  sn = block_reduce_sum(sn, scr);
  sp = block_reduce_sum(sp, scr);

  if (tid == 0) {
    const float LN2 = 0.6931471805599453f;
    const float l2n = (sn > 0.0f) ? (mn + log2f(sn)) : NEGINF;
    const float l2p = (sp > 0.0f) ? (mp + log2f(sp)) : NEGINF;
    const float z = l2n + l2p;  // base-2 domain total
    // softplus(ln2 * z) = ln2 * (max(z,0) + log2(1 + 2^-|z|)); z=-inf -> 0
    const float per = LN2 * (fmaxf(z, 0.0f) + log2f(1.0f + exp2f(-fabsf(z))));
    row_loss[row] = per;
  }
}

// ---- deterministic final mean over rows (also initializes d_out) ----
__global__ __launch_bounds__(TPB) void lsep_final_kernel(
    const float* __restrict__ row_loss, float* __restrict__ out, int nrows) {
  __shared__ float scr[TPB / 32 * 2];
  float s = 0.0f;
  for (int i = threadIdx.x; i < nrows; i += TPB) s += row_loss[i];
  s = block_reduce_sum(s, scr);
  if (threadIdx.x == 0) out[0] = s / (float)nrows;
}

extern "C" void kernel_launch(void* const* d_in, const int* in_sizes, int n_in,
                              void* d_out, int out_size, void* d_ws, size_t ws_size,
                              hipStream_t stream) {
  (void)n_in; (void)out_size; (void)ws_size;
  const float* x  = (const float*)d_in[0];
  const int*   tg = (const int*)d_in[1];
  float* row_loss = (float*)d_ws;           // nrows floats of scratch
  float* out      = (float*)d_out;
  const int nrows = in_sizes[0] / CDIM;     // 4096

  lsep_row_kernel<<<dim3(nrows), dim3(TPB), 0, stream>>>(x, tg, row_loss);
  lsep_final_kernel<<<dim3(1), dim3(TPB), 0, stream>>>(row_loss, out, nrows);
}